// LocalScoreModule_11553462026625
// MI455X (gfx1250) — compile-verified
//
#include <hip/hip_runtime.h>

// MI455X / gfx1250, wave32. Distance-as-GEMM via V_WMMA_F32_16X16X4_F32,
// separable 5x5 box filter in LDS, online-softmax partials + final reduce.

typedef float v2f __attribute__((ext_vector_type(2)));
typedef float v8f __attribute__((ext_vector_type(8)));

#define HW      1024      // 32*32 pixels
#define BQ      8         // query batch
#define PITCH   33        // LDS row pitch (bank skew)
#define PLANE   264       // 8 staged rows * PITCH
#define NPLANES 128       // 8 b * 16 n
#define NCHUNK  64        // n-chunks of 64 images
#define NT      4         // 16-image tiles per chunk
#define RSTRIP  4         // output rows per strip (staged rows = RSTRIP+4 = 8)

// ---- Pass A: xn[b,p] = sum_c x^2 ------------------------------------------
__global__ __launch_bounds__(256) void lsm_xn_kernel(const float* __restrict__ x,
                                                     float* __restrict__ xn) {
  int i = blockIdx.x * 256 + threadIdx.x;        // 8192 items
  int b = i >> 10, p = i & 1023;
  const float* xp = x + (size_t)b * 3072 + p;
  float v0 = xp[0], v1 = xp[1024], v2 = xp[2048];
  xn[i] = v0 * v0 + v1 * v1 + v2 * v2;
}

// ---- Pass B: per-pixel WMMA distances -> 5x5 window sum -> online softmax --
__global__ __launch_bounds__(256) void lsm_score_kernel(
    const float* __restrict__ tptr, const float* __restrict__ x,
    const float* __restrict__ images, const float* __restrict__ xn,
    float* __restrict__ Pm, float* __restrict__ PS,
    float* __restrict__ PT0, float* __restrict__ PT1, float* __restrict__ PT2) {
  __shared__ float ns[NPLANES * PLANE];          // 135,168 B of LDS
  const int tid  = threadIdx.x;
  const int lane = tid & 31;
  const int wave = tid >> 5;                     // 8 waves; wave == staged row == b
  const int chunk = blockIdx.x;                  // 0..63
  const int r0    = blockIdx.y * RSTRIP;         // 0..28

  const float tt    = tptr[0];
  const float s     = 1.0f - __expf(-2.0f * tt); // bt^2
  const float alpha = 1.0f / (2.0f * s);
  const float at    = sqrtf(1.0f - s);
  const float at2   = 1.0f - s;
  const float m2at  = -2.0f * at;

  const int  nl = lane & 15;
  const bool hi = lane >= 16;

  float M[RSTRIP], S[RSTRIP], T0[RSTRIP], T1[RSTRIP], T2[RSTRIP];
#pragma unroll
  for (int r = 0; r < RSTRIP; ++r) {
    M[r] = __builtin_inff(); S[r] = 0.f; T0[r] = 0.f; T1[r] = 0.f; T2[r] = 0.f;
  }

  for (int tile = 0; tile < NT; ++tile) {
    const int n0 = (chunk * NT + tile) * 16;
    __syncthreads();                             // prev tile's consumers done

    // -- per-pixel WMMA: normsq[16b x 16n] = A(16x4) * B(4x16) + xn ---------
    // A row b : [x0, x1, x2, 1]
    // B col n : [-2at*i0, -2at*i1, -2at*i2, at^2*(i0^2+i1^2+i2^2)]
    const int gr = r0 - 2 + wave;                // global row (uniform per wave)
    if (gr >= 0 && gr < 32) {
      for (int col = 0; col < 32; ++col) {
        const int gp = gr * 32 + col;
        float ax = 0.f, ay = 0.f;
        if (nl < BQ) {
          const float* xp = x + (size_t)nl * 3072 + gp;
          if (!hi) { ax = xp[0];    ay = xp[1024]; }  // K0,K1
          else     { ax = xp[2048]; ay = 1.0f;     }  // K2,K3
        }
        const float* ip = images + (size_t)(n0 + nl) * 3072 + gp;
        float bx, by;
        if (!hi) { bx = m2at * ip[0]; by = m2at * ip[1024]; }
        else {
          float i0 = ip[0], i1 = ip[1024], i2 = ip[2048];
          bx = m2at * i2;
          by = at2 * (i0 * i0 + i1 * i1 + i2 * i2);
        }
        v2f a;  a.x = ax;  a.y = ay;
        v2f bf; bf.x = bx; bf.y = by;
        v8f c = {};
        v8f d = __builtin_amdgcn_wmma_f32_16x16x4_f32(
            false, a, false, bf, (short)0, c, false, false);
        if (lane < 16) {                         // d[j] lanes0-15 = (b=j, n=lane)
#pragma unroll
          for (int j = 0; j < BQ; ++j)
            ns[(j * 16 + nl) * PLANE + wave * PITCH + col] =
                d[j] + xn[j * 1024 + gp];
        }
      }
    } else {                                     // zero-pad rows outside image
      if (lane < 16)
        for (int col = 0; col < 32; ++col)
#pragma unroll
          for (int j = 0; j < BQ; ++j)
            ns[(j * 16 + nl) * PLANE + wave * PITCH + col] = 0.f;
    }
    __syncthreads();

    // -- horizontal 5-tap zero-padded window sum, in place ------------------
    for (int q = 0; q < 4; ++q) {
      int pair = tid + q * 256;                  // 128 planes * 8 rows
      float* base = ns + (pair >> 3) * PLANE + (pair & 7) * PITCH;
      float w0 = 0.f, w1 = 0.f, w2 = 0.f, w3 = 0.f, w4 = 0.f;
      for (int cc = 0; cc < 34; ++cc) {
        float v = (cc < 32) ? base[cc] : 0.f;
        w0 = w1; w1 = w2; w2 = w3; w3 = w4; w4 = v;
        if (cc >= 2) base[cc - 2] = w0 + w1 + w2 + w3 + w4;
      }
    }
    __syncthreads();

    // -- vertical 5-tap: valid centers rows 2..5 (global r0..r0+3) ----------
    for (int q = 0; q < 16; ++q) {
      int pair = tid + q * 256;                  // 128 planes * 32 cols
      float* base = ns + (pair >> 5) * PLANE + (pair & 31);
      float w0 = 0.f, w1 = 0.f, w2 = 0.f, w3 = 0.f, w4 = 0.f;
      for (int r = 0; r < 8; ++r) {
        float v = base[r * PITCH];
        w0 = w1; w1 = w2; w2 = w3; w3 = w4; w4 = v;
        if (r >= 4) base[(r - 2) * PITCH] = w0 + w1 + w2 + w3 + w4;
      }
    }
    __syncthreads();

    // -- online-softmax accumulation over this tile's 16 images -------------
    {
      const int bq = wave, col = lane;
#pragma unroll
      for (int row = 0; row < RSTRIP; ++row) {
        const int rl = row + 2;
        float mt = __builtin_inff();
        for (int n = 0; n < 16; ++n)
          mt = fminf(mt, ns[(bq * 16 + n) * PLANE + rl * PITCH + col]);
        float Mn = fminf(M[row], mt);
        float rs = __expf((Mn - M[row]) * alpha);   // M=inf first tile -> 0
        float Sv = S[row] * rs;
        float a0 = T0[row] * rs, a1 = T1[row] * rs, a2 = T2[row] * rs;
        const int gp = (r0 + row) * 32 + col;
        for (int n = 0; n < 16; ++n) {
          float w = ns[(bq * 16 + n) * PLANE + rl * PITCH + col];
          float e = __expf((Mn - w) * alpha);
          const float* ip = images + (size_t)(n0 + n) * 3072 + gp;
          Sv += e; a0 += e * ip[0]; a1 += e * ip[1024]; a2 += e * ip[2048];
        }
        M[row] = Mn; S[row] = Sv; T0[row] = a0; T1[row] = a1; T2[row] = a2;
      }
    }
  }

  // -- write per-chunk partials ---------------------------------------------
#pragma unroll
  for (int row = 0; row < RSTRIP; ++row) {
    size_t idx = ((size_t)wave * NCHUNK + chunk) * HW + (size_t)(r0 + row) * 32 + lane;
    Pm[idx] = M[row]; PS[idx] = S[row];
    PT0[idx] = T0[row]; PT1[idx] = T1[row]; PT2[idx] = T2[row];
  }
}

// ---- Pass C: combine chunk partials, final output -------------------------
__global__ __launch_bounds__(256) void lsm_reduce_kernel(
    const float* __restrict__ tptr, const float* __restrict__ x,
    const float* __restrict__ Pm, const float* __restrict__ PS,
    const float* __restrict__ PT0, const float* __restrict__ PT1,
    const float* __restrict__ PT2, float* __restrict__ out) {
  int i = blockIdx.x * 256 + threadIdx.x;        // 8192 = (b,p)
  int b = i >> 10, p = i & 1023;
  const float tt    = tptr[0];
  const float s     = 1.0f - __expf(-2.0f * tt);
  const float alpha = 1.0f / (2.0f * s);
  const float at    = sqrtf(1.0f - s);
  float M = __builtin_inff(), S = 0.f, t0 = 0.f, t1 = 0.f, t2 = 0.f;
  for (int ch = 0; ch < NCHUNK; ++ch) {
    size_t idx = ((size_t)b * NCHUNK + ch) * HW + p;
    float pm = Pm[idx];
    float Mn = fminf(M, pm);
    float f0 = __expf((Mn - M) * alpha);
    float f1 = __expf((Mn - pm) * alpha);
    S  = S  * f0 + PS[idx]  * f1;
    t0 = t0 * f0 + PT0[idx] * f1;
    t1 = t1 * f0 + PT1[idx] * f1;
    t2 = t2 * f0 + PT2[idx] * f1;
    M = Mn;
  }
  float inv  = 1.0f / S;
  float ibt2 = 1.0f / s;
  const float* xp = x + (size_t)b * 3072 + p;
  float* op = out + (size_t)b * 3072 + p;
  op[0]    = -(xp[0]    - at * t0 * inv) * ibt2;
  op[1024] = -(xp[1024] - at * t1 * inv) * ibt2;
  op[2048] = -(xp[2048] - at * t2 * inv) * ibt2;
}

extern "C" void kernel_launch(void* const* d_in, const int* in_sizes, int n_in,
                              void* d_out, int out_size, void* d_ws, size_t ws_size,
                              hipStream_t stream) {
  (void)in_sizes; (void)n_in; (void)out_size; (void)ws_size;
  const float* t      = (const float*)d_in[0];
  const float* x      = (const float*)d_in[1];
  const float* images = (const float*)d_in[2];
  // d_in[3] is k; the window (k=5) is baked into the separable filter.

  float* ws  = (float*)d_ws;
  float* xn  = ws;                   // 8192 floats
  float* Pm  = ws + 8192;            // 5 partial arrays, 8*64*1024 floats each
  float* PS  = Pm  + 8 * NCHUNK * HW;
  float* PT0 = PS  + 8 * NCHUNK * HW;
  float* PT1 = PT0 + 8 * NCHUNK * HW;
  float* PT2 = PT1 + 8 * NCHUNK * HW;

  lsm_xn_kernel<<<32, 256, 0, stream>>>(x, xn);
  lsm_score_kernel<<<dim3(NCHUNK, 8), 256, 0, stream>>>(t, x, images, xn,
                                                        Pm, PS, PT0, PT1, PT2);
  lsm_reduce_kernel<<<32, 256, 0, stream>>>(t, x, Pm, PS, PT0, PT1, PT2,
                                            (float*)d_out);
}